// GAConv1D_83081847374309
// MI455X (gfx1250) — compile-verified
//
#include <hip/hip_runtime.h>

#define B_ 16
#define C_ 64
#define S_ 16384
#define O_ 128
#define K_ 9

#define STRXT 68         // LDS row stride (floats) for transposed x tile [s][c]
#define STILE 64         // S columns per x tile
#define TILES_PER_BLOCK 4

typedef float v2f __attribute__((ext_vector_type(2)));
typedef float v8f __attribute__((ext_vector_type(8)));

// ---------------------------------------------------------------------------
// Kernel 1: collapse rotation params to W[o][c].
// Sum of all 9 rotation-matrix entries = 3 + 4(bc+bd+cd - b^2 - c^2 - d^2);
// the a = phi*cos(theta) terms cancel. W = mean_j sum_{K,k} R = 9 + (4/3)*sum_K s^2*g.
// ---------------------------------------------------------------------------
__global__ void ga_weights_kernel(const float* __restrict__ phi,
                                  const float* __restrict__ theta,
                                  const float* __restrict__ pb,
                                  const float* __restrict__ pc,
                                  const float* __restrict__ pd,
                                  float* __restrict__ W) {
  int idx = blockIdx.x * blockDim.x + threadIdx.x;   // idx = o*C + c
  if (idx >= O_ * C_) return;
  int base = idx * K_;
  float acc = 0.0f;
#pragma unroll
  for (int k = 0; k < K_; ++k) {
    float ph = phi[base + k];
    float th = theta[base + k];
    float b = pb[base + k];
    float c = pc[base + k];
    float d = pd[base + k];
    float s = ph * sinf(th);
    float g = b * c + b * d + c * d - b * b - c * c - d * d;
    acc = fmaf(s * s, g, acc);
  }
  W[idx] = 9.0f + (4.0f / 3.0f) * acc;
}

// ---------------------------------------------------------------------------
// CDNA5 async global->LDS copies (VGLOBAL encoding, tracked by ASYNCcnt).
// VDST = per-lane LDS byte address, VADDR = per-lane 64-bit global address.
// ---------------------------------------------------------------------------
__device__ __forceinline__ void async_load_lds_b32(unsigned lds_off,
                                                   const void* gptr) {
  asm volatile("global_load_async_to_lds_b32 %0, %1, off"
               :
               : "v"(lds_off), "v"((unsigned long long)(uintptr_t)gptr)
               : "memory");
}

__device__ __forceinline__ void wait_async0() {
#if __has_builtin(__builtin_amdgcn_s_wait_asynccnt)
  __builtin_amdgcn_s_wait_asynccnt(0);
#else
  asm volatile("s_wait_asynccnt 0" ::: "memory");
#endif
}

// Stage x[b, 0..63, s0..s0+63] TRANSPOSED into LDS: buf[s*STRXT + c].
// Each thread owns one s column (s = tid&63) and walks c in steps of 4.
// Global side: 32 consecutive s per wave -> fully coalesced 128B lines.
__device__ __forceinline__ void stage_tile_async(const float* __restrict__ x,
                                                 float* __restrict__ buf,
                                                 int tid, int b, int s0) {
  const int s  = tid & 63;
  const int c0 = tid >> 6;                 // 0..3
  const float* gp = &x[((size_t)b * C_ + c0) * S_ + s0 + s];
  const float* lp = &buf[s * STRXT + c0];
#pragma unroll
  for (int p = 0; p < C_ / 4; ++p) {
    async_load_lds_b32((unsigned)(unsigned long long)(uintptr_t)(lp + 4 * p),
                       gp + (size_t)(4 * p) * S_);
  }
}

// ---------------------------------------------------------------------------
// Kernel 2: out[b,o,s] = sum_c W[o,c] * x[b,c,s] via V_WMMA_F32_16X16X4_F32.
// Block = 256 threads = 8 waves. Block owns (b, 4 consecutive 64-wide s tiles).
// A fragments (16 k-steps x v2f = 32 VGPRs) loaded once per wave straight from
// the L2-resident W array. x tiles double-buffered in LDS (transposed layout)
// via global_load_async_to_lds_b32; B fragments are single aligned ds_load_b64s
// that land directly in the even-aligned pairs WMMA requires (no marshalling).
// ---------------------------------------------------------------------------
__global__ __launch_bounds__(256) void ga_gemm_kernel(const float* __restrict__ x,
                                                      const float* __restrict__ W,
                                                      float* __restrict__ out) {
  __shared__ float Xlds[2][STILE * STRXT];   // 2 x 17408 B, transposed [s][c]

  const int tid  = threadIdx.x;
  const int lane = tid & 31;
  const int wave = tid >> 5;                 // 0..7 -> O tile
  const int b    = blockIdx.y;
  const int s_block0 = blockIdx.x * (STILE * TILES_PER_BLOCK);

  // ---- Kick off async staging of the first x tile immediately ----
  stage_tile_async(x, Xlds[0], tid, b, s_block0);

  const int o_base = wave * 16;
  const int m      = lane & 15;              // A row / D column
  const int khalf  = (lane >> 4) * 2;        // K sub-offset per half-wave

  // ---- A fragments straight from global W (overlaps with async staging) ----
  v2f aFrag[16];
#pragma unroll
  for (int ks = 0; ks < 16; ++ks) {
    aFrag[ks] = *reinterpret_cast<const v2f*>(
        &W[(o_base + m) * C_ + ks * 4 + khalf]);   // 8B aligned (khalf even)
  }

  wait_async0();                             // tile 0 resident in LDS
  __syncthreads();                           // publish tile 0 to all waves

  for (int t = 0; t < TILES_PER_BLOCK; ++t) {
    const int s0 = s_block0 + t * STILE;
    const float* __restrict__ Xbuf = Xlds[t & 1];

    // Stream the next tile into the other buffer while this one computes.
    if (t + 1 < TILES_PER_BLOCK)
      stage_tile_async(x, Xlds[(t + 1) & 1], tid, b, s0 + STILE);

    // ---- 4 x (16x16) output sub-tiles, each 16 chained WMMAs over C=64 ----
#pragma unroll
    for (int ssub = 0; ssub < 4; ++ssub) {
      const int sl = ssub * 16 + m;          // local S column for B/D
      v8f acc = {0.f, 0.f, 0.f, 0.f, 0.f, 0.f, 0.f, 0.f};
#pragma unroll
      for (int ks = 0; ks < 16; ++ks) {
        // B fragment: (x[c][sl], x[c+1][sl]) = contiguous 8B in transposed LDS
        v2f bFrag = *reinterpret_cast<const v2f*>(
            &Xbuf[sl * STRXT + ks * 4 + khalf]);
        acc = __builtin_amdgcn_wmma_f32_16x16x4_f32(
            /*neg_a=*/false, aFrag[ks], /*neg_b=*/false, bFrag,
            /*c_mod=*/(short)0, acc, /*reuse_a=*/false, /*reuse_b=*/false);
      }
      // ---- Store D: VGPR r -> row o_base + r + 8*(lane>>4), col sl ----
#pragma unroll
      for (int r = 0; r < 8; ++r) {
        int orow = o_base + r + 8 * (lane >> 4);
        out[((size_t)b * O_ + orow) * S_ + s0 + sl] = acc[r];
      }
    }

    // Next tile fully landed in LDS, then publish to all waves.
    wait_async0();
    __syncthreads();
  }
}

// ---------------------------------------------------------------------------
extern "C" void kernel_launch(void* const* d_in, const int* in_sizes, int n_in,
                              void* d_out, int out_size, void* d_ws, size_t ws_size,
                              hipStream_t stream) {
  (void)in_sizes; (void)n_in; (void)out_size; (void)ws_size;
  const float* x     = (const float*)d_in[0];
  const float* phi   = (const float*)d_in[1];
  const float* theta = (const float*)d_in[2];
  const float* pb    = (const float*)d_in[3];
  const float* pc    = (const float*)d_in[4];
  const float* pd    = (const float*)d_in[5];
  float* W   = (float*)d_ws;    // 128*64*4 = 32 KB scratch
  float* out = (float*)d_out;

  ga_weights_kernel<<<dim3((O_ * C_ + 255) / 256), dim3(256), 0, stream>>>(
      phi, theta, pb, pc, pd, W);

  dim3 grid(S_ / (STILE * TILES_PER_BLOCK), B_);   // (64, 16)
  ga_gemm_kernel<<<grid, dim3(256), 0, stream>>>(x, W, out);
}